// GDMLTorchPredict_10342281249142
// MI455X (gfx1250) — compile-verified
//
#include <hip/hip_runtime.h>
#include <math.h>

// Problem constants (from reference)
#define M_BATCH 64
#define N_ATOMS 24
#define T_TRAIN 8000
#define D_DESC  276
#define Q_CONST  0.22360679775f   // sqrt(5)/SIG, SIG=10
#define K0_CONST 0.01666666667f   // 5/(3*SIG^2)
#define INVQ_CONST 4.472135955f   // SIG/sqrt(5)

typedef __attribute__((ext_vector_type(2))) float v2f;
typedef __attribute__((ext_vector_type(8))) float v8f;
typedef __attribute__((ext_vector_type(4))) unsigned int v4u;
typedef __attribute__((ext_vector_type(8))) int v8i;
typedef __attribute__((ext_vector_type(4))) int v4i;

// Workspace layout (float offsets)
#define OFF_XS     0u                 // 64*276        = 17664
#define OFF_PV     17664u             // 64*276*3      = 52992
#define OFF_XN2    70656u             // 64
#define OFF_C      70720u             // 8000
#define OFF_TN2    78720u             // 8000
#define OFF_ESACC  86720u             // 64
#define OFF_SUMA   86784u             // 64
#define OFF_A      86848u             // 64*8000      = 512000
#define OFF_B      598848u            // 512000
#define OFF_FSX    1110848u           // 64*276       = 17664
// total 1128512 floats = ~4.52 MB

// Pass-2 K blocking: 25 chunks of 320, TDM slabs of 32 rows x 16 cols
#define KCHUNK 320
#define KSLAB  32
#define NSLAB  (KCHUNK / KSLAB)   // 10

__device__ __forceinline__ float half_reduce16(float v) {
    // sum across each 16-lane half of a wave32 (xor stays within halves)
    v += __shfl_xor(v, 1, 32);
    v += __shfl_xor(v, 2, 32);
    v += __shfl_xor(v, 4, 32);
    v += __shfl_xor(v, 8, 32);
    return v;
}

// ---------------------------------------------------------------------------
// TDM: async-load a KSLAB x 16 f32 tile (row stride D_DESC) into LDS.
// D# per CDNA5 ISA ch.8: group0 = {count=1, lds_addr, global_addr, type=2},
// group1 = {data_size=4B, tensor_dim0=valid_cols (OOB cols read as 0 ->
// free tail padding for D=276), tensor_dim1=KSLAB, tile=16 x KSLAB,
// dim0_stride=276}.  Tracked with TENSORcnt.
// This toolchain exposes the 6-arg builtin:
//   (uint32x4 g0, int32x8 g1, int32x4 g2, int32x4 g3, int32x8 g4, i32 cpol)
// ---------------------------------------------------------------------------
__device__ __forceinline__ void tdm_load_tile(const float* gsrc,
                                              unsigned lds_byte,
                                              unsigned valid_cols) {
    unsigned long long ga = (unsigned long long)(uintptr_t)gsrc;
    v4u g0;
    g0.x = 1u;                                                 // count=1
    g0.y = lds_byte;                                           // lds_addr
    g0.z = (unsigned)(ga & 0xFFFFFFFFu);                       // global_addr lo
    g0.w = (unsigned)((ga >> 32) & 0x01FFFFFFu) | (2u << 30);  // ga hi | type=2
    v8i g1;
    g1[0] = (int)(2u << 16);                        // data_size = 4 bytes
    g1[1] = (int)((valid_cols & 0xFFFFu) << 16);    // tensor_dim0[15:0]
    g1[2] = (int)((unsigned)KSLAB << 16);           // dim0 hi=0 | tensor_dim1 lo
    g1[3] = (int)(16u << 16);                       // dim1 hi=0 | tile_dim0=16
    g1[4] = (int)KSLAB;                             // tile_dim1 | tile_dim2=0
    g1[5] = (int)D_DESC;                            // tensor_dim0_stride lo
    g1[6] = 0;                                      // stride hi | dim1_stride lo
    g1[7] = 0;
    v4i z4 = {0, 0, 0, 0};                          // groups 2/3 unused (2D)
    v8i z8 = {0, 0, 0, 0, 0, 0, 0, 0};
    __builtin_amdgcn_tensor_load_to_lds(g0, g1, z4, z4, z8, 0);
}

// ---------------------------------------------------------------------------
// Kernel 0: descriptors xs = 1/dist and pair vectors.  grid=64, block=288
// ---------------------------------------------------------------------------
__global__ void k_desc(const float* __restrict__ Rs,
                       float* __restrict__ xs, float* __restrict__ pv) {
    int m = blockIdx.x;
    int p = threadIdx.x;
    if (p >= D_DESC) return;
    // tril_indices(24, -1) ordering: for i in 1..23: for j in 0..i-1
    int i = 1, base = 0;
    while (base + i <= p) { base += i; ++i; }
    int j = p - base;
    const float* rm = Rs + m * N_ATOMS * 3;
    float dx = rm[i*3+0] - rm[j*3+0];
    float dy = rm[i*3+1] - rm[j*3+1];
    float dz = rm[i*3+2] - rm[j*3+2];
    float r  = sqrtf(dx*dx + dy*dy + dz*dz);
    xs[m * D_DESC + p] = 1.0f / r;
    float* o = pv + (m * D_DESC + p) * 3;
    o[0] = dx; o[1] = dy; o[2] = dz;
}

// ---------------------------------------------------------------------------
// Kernel 1: ||xs[m]||^2 ; also zero Es/rowsum accumulators. grid=64, block=64
// ---------------------------------------------------------------------------
__global__ void k_xn2(const float* __restrict__ xs, float* __restrict__ xn2,
                      float* __restrict__ esacc, float* __restrict__ suma) {
    __shared__ float red[64];
    int m = blockIdx.x, t = threadIdx.x;
    if (m == 0) { esacc[t] = 0.0f; suma[t] = 0.0f; }
    float s = 0.0f;
    for (int d = t; d < D_DESC; d += 64) {
        float v = xs[m * D_DESC + d];
        s += v * v;
    }
    red[t] = s; __syncthreads();
    for (int w = 32; w > 0; w >>= 1) {
        if (t < w) red[t] += red[t + w];
        __syncthreads();
    }
    if (t == 0) xn2[m] = red[0];
}

// ---------------------------------------------------------------------------
// Kernel 2: c[t]=xs_t.Jx_t, tn2[t]=||xs_t||^2; also zero Fs_x. grid=8000, blk=64
// ---------------------------------------------------------------------------
__global__ void k_train(const float* __restrict__ xt, const float* __restrict__ ja,
                        float* __restrict__ c, float* __restrict__ tn2,
                        float* __restrict__ fsx) {
    __shared__ float r1[64], r2[64];
    int t = blockIdx.x, l = threadIdx.x;
    if (t < 276) fsx[t * 64 + l] = 0.0f;   // 276*64 == 64*276 floats
    float sc = 0.0f, sn = 0.0f;
    const float* x = xt + (size_t)t * D_DESC;
    const float* j = ja + (size_t)t * D_DESC;
    for (int d = l; d < D_DESC; d += 64) {
        float xv = x[d];
        sc += xv * j[d];
        sn += xv * xv;
    }
    r1[l] = sc; r2[l] = sn; __syncthreads();
    for (int w = 32; w > 0; w >>= 1) {
        if (l < w) { r1[l] += r1[l + w]; r2[l] += r2[l + w]; }
        __syncthreads();
    }
    if (l == 0) { c[t] = r1[0]; tn2[t] = r2[0]; }
}

// ---------------------------------------------------------------------------
// Kernel 3: pass 1.  Two f32 WMMA chains over K=D=276 computing
//   S1 = xs @ xs_train^T, S2 = xs @ Jx^T   for one 16x16 (m,t) tile per wave,
// then elementwise Matern-5/2 math; writes a,b; atomic partials for Es, suma.
// grid=500 (t tiles), block=128 (wave w = m tile w)
// ---------------------------------------------------------------------------
__global__ void __launch_bounds__(128)
k_pass1(const float* __restrict__ xs, const float* __restrict__ xt,
        const float* __restrict__ ja, const float* __restrict__ xn2,
        const float* __restrict__ c, const float* __restrict__ tn2,
        float* __restrict__ a, float* __restrict__ b,
        float* __restrict__ esacc, float* __restrict__ suma) {
    const int lane  = threadIdx.x & 31;
    const int mtile = threadIdx.x >> 5;     // 0..3
    const int ttile = blockIdx.x;           // 0..499
    const int hi = lane >> 4;               // lane half
    const int lo = lane & 15;

    v8f s1 = {}; v8f s2 = {};
    // A frag (16x4 f32): lane row M=lo, K offset 2*hi -> contiguous float2
    const float* arow = xs + (size_t)(mtile * 16 + lo) * D_DESC + 2 * hi;
    // B frag (4x16 f32): lane col N=lo (t index), K offset 2*hi; B[k][n]=xt[t=n][d=k]
    const float* brow = xt + (size_t)(ttile * 16 + lo) * D_DESC + 2 * hi;
    const float* jrow = ja + (size_t)(ttile * 16 + lo) * D_DESC + 2 * hi;

    for (int k = 0; k < D_DESC; k += 4) {   // 276 = 69*4, no tail
        v2f av = *(const v2f*)(arow + k);
        v2f bx = *(const v2f*)(brow + k);
        v2f bj = *(const v2f*)(jrow + k);
        s1 = __builtin_amdgcn_wmma_f32_16x16x4_f32(false, av, false, bx,
                                                   (short)0, s1, false, false);
        s2 = __builtin_amdgcn_wmma_f32_16x16x4_f32(false, av, false, bj,
                                                   (short)0, s2, false, false);
    }

    const int t = ttile * 16 + lo;          // C layout: N = lane&15
    const float cv = c[t], tn = tn2[t];
#pragma unroll
    for (int v = 0; v < 8; ++v) {
        const int m = mtile * 16 + v + 8 * hi;   // C layout: M = v + 8*half
        float d2 = xn2[m] - 2.0f * s1[v] + tn;
        d2 = fmaxf(d2, 0.0f);
        float xd = Q_CONST * sqrtf(d2);                   // x_dists
        float ex = K0_CONST * __expf(-xd);                // exp_xs
        float dv = Q_CONST * (s2[v] - cv);                // dot
        float e1 = ex * (1.0f + xd);                      // exp1
        float av_ = ex * dv;                              // "a" weights
        a[(size_t)m * T_TRAIN + t] = av_;
        b[(size_t)m * T_TRAIN + t] = e1;
        float es = half_reduce16(e1 * dv);                // Es partials
        float sa = half_reduce16(av_);                    // rowsum(a) partials
        if (lo == 0) {
            atomicAdd(&esacc[m], es);
            atomicAdd(&suma[m], sa);
        }
    }
}

// ---------------------------------------------------------------------------
// Kernel 4: pass 2.  Two f32 WMMA chains over K computing
//   F1 = a @ xs_train, F2 = b @ Jx_alphas, accumulating -q*F1 - F2 into Fs_x.
// B slabs (shared by all 4 waves of the block) are staged into LDS by the
// Tensor Data Mover, double-buffered, synced via TENSORcnt + block barrier.
// grid=(18 d-tiles, 25 k-chunks), block=128 (wave w = m tile w)
// ---------------------------------------------------------------------------
__global__ void __launch_bounds__(128)
k_pass2(const float* __restrict__ a, const float* __restrict__ b,
        const float* __restrict__ xt, const float* __restrict__ ja,
        float* __restrict__ fsx) {
    __shared__ float sbx[2][KSLAB * 16];   // xs_train slab (2.0 KB x2)
    __shared__ float sbj[2][KSLAB * 16];   // Jx_alphas slab
    const int lane  = threadIdx.x & 31;
    const int mtile = threadIdx.x >> 5;     // 0..3
    const int dtile = blockIdx.x;           // 0..17
    const int kc    = blockIdx.y;           // 0..24
    const int hi = lane >> 4, lo = lane & 15;
    const int d  = dtile * 16 + lo;
    const unsigned validc = (unsigned)(D_DESC - dtile * 16); // OOB cols -> 0
    const int kbeg = kc * KCHUNK;
    const bool wave0 = (threadIdx.x < 32);  // wave-uniform: TDM ignores EXEC,
                                            // so gate by branch, one wave issues

    if (wave0) {
        tdm_load_tile(xt + (size_t)kbeg * D_DESC + dtile * 16,
                      (unsigned)(uintptr_t)&sbx[0][0], validc);
        tdm_load_tile(ja + (size_t)kbeg * D_DESC + dtile * 16,
                      (unsigned)(uintptr_t)&sbj[0][0], validc);
    }

    v8f f1 = {}; v8f f2 = {};
    const float* arow = a + (size_t)(mtile * 16 + lo) * T_TRAIN + 2 * hi;
    const float* brow = b + (size_t)(mtile * 16 + lo) * T_TRAIN + 2 * hi;

    for (int s = 0; s < NSLAB; ++s) {
        if (wave0) __builtin_amdgcn_s_wait_tensorcnt(0);
        __syncthreads();                          // slab s resident in LDS
        if (wave0 && (s + 1) < NSLAB) {           // prefetch slab s+1 into other buf
            const int knext = kbeg + (s + 1) * KSLAB;
            const int nb = (s + 1) & 1;
            tdm_load_tile(xt + (size_t)knext * D_DESC + dtile * 16,
                          (unsigned)(uintptr_t)&sbx[nb][0], validc);
            tdm_load_tile(ja + (size_t)knext * D_DESC + dtile * 16,
                          (unsigned)(uintptr_t)&sbj[nb][0], validc);
        }
        const float* lx = &sbx[s & 1][0];
        const float* lj = &sbj[s & 1][0];
        const int kglob = kbeg + s * KSLAB;
#pragma unroll 4
        for (int ks = 0; ks < KSLAB; ks += 4) {
            v2f av = *(const v2f*)(arow + kglob + ks);
            v2f bv = *(const v2f*)(brow + kglob + ks);
            const int kk = ks + 2 * hi;
            v2f bx, bj;
            bx.x = lx[kk * 16 + lo];
            bx.y = lx[(kk + 1) * 16 + lo];
            bj.x = lj[kk * 16 + lo];
            bj.y = lj[(kk + 1) * 16 + lo];
            f1 = __builtin_amdgcn_wmma_f32_16x16x4_f32(false, av, false, bx,
                                                       (short)0, f1, false, false);
            f2 = __builtin_amdgcn_wmma_f32_16x16x4_f32(false, bv, false, bj,
                                                       (short)0, f2, false, false);
        }
        __syncthreads();   // all waves done with buf (s&1) before TDM s+2 reuses it
    }

    if (d < D_DESC) {
#pragma unroll
        for (int v = 0; v < 8; ++v) {
            const int m = mtile * 16 + v + 8 * hi;
            atomicAdd(&fsx[(size_t)m * D_DESC + d], -Q_CONST * f1[v] - f2[v]);
        }
    }
}

// ---------------------------------------------------------------------------
// Kernel 5: finalize.  Fs_x += q*xs*rowsum(a); chain rule *xs^3; scatter to
// atoms; Es = esacc/q.  grid=64 (m), block=96 (72 = 24 atoms x 3 axes used)
// ---------------------------------------------------------------------------
__global__ void k_final(const float* __restrict__ fsx, const float* __restrict__ xs,
                        const float* __restrict__ pv, const float* __restrict__ suma,
                        const float* __restrict__ esacc, float* __restrict__ out) {
    const int m = blockIdx.x;
    const int tid = threadIdx.x;
    if (tid < N_ATOMS * 3) {
        const int n = tid / 3, ax = tid % 3;
        const float sm = suma[m];
        float acc = 0.0f;
        int i = 1, j = 0;
        for (int p = 0; p < D_DESC; ++p) {
            const float x = xs[m * D_DESC + p];
            const float s = (fsx[m * D_DESC + p] + Q_CONST * x * sm) * x * x * x;
            const float comp = s * pv[(m * D_DESC + p) * 3 + ax];
            if (j == n) acc += comp;     // Fs[:, j] += scaled
            if (i == n) acc -= comp;     // Fs[:, i] -= scaled
            ++j; if (j == i) { ++i; j = 0; }
        }
        out[M_BATCH + (m * N_ATOMS + n) * 3 + ax] = acc;  // STD = 1
    }
    if (tid == N_ATOMS * 3) {
        out[m] = esacc[m] * INVQ_CONST;   // Es = sum(exp1*dot)/q + C, C=0
    }
}

// ---------------------------------------------------------------------------
extern "C" void kernel_launch(void* const* d_in, const int* in_sizes, int n_in,
                              void* d_out, int out_size, void* d_ws, size_t ws_size,
                              hipStream_t stream) {
    (void)in_sizes; (void)n_in; (void)out_size; (void)ws_size;
    const float* Rs = (const float*)d_in[0];   // [64,24,3]
    const float* xt = (const float*)d_in[1];   // [8000,276]
    const float* ja = (const float*)d_in[2];   // [8000,276]
    float* out = (float*)d_out;                // [64] Es ++ [64,24,3] Fs
    float* ws  = (float*)d_ws;

    float* xs    = ws + OFF_XS;
    float* pv    = ws + OFF_PV;
    float* xn2   = ws + OFF_XN2;
    float* c     = ws + OFF_C;
    float* tn2   = ws + OFF_TN2;
    float* esacc = ws + OFF_ESACC;
    float* suma  = ws + OFF_SUMA;
    float* a     = ws + OFF_A;
    float* b     = ws + OFF_B;
    float* fsx   = ws + OFF_FSX;

    k_desc <<<M_BATCH, 288, 0, stream>>>(Rs, xs, pv);
    k_xn2  <<<M_BATCH, 64, 0, stream>>>(xs, xn2, esacc, suma);
    k_train<<<T_TRAIN, 64, 0, stream>>>(xt, ja, c, tn2, fsx);
    k_pass1<<<T_TRAIN / 16, 128, 0, stream>>>(xs, xt, ja, xn2, c, tn2,
                                              a, b, esacc, suma);
    k_pass2<<<dim3(18, 25), 128, 0, stream>>>(a, b, xt, ja, fsx);
    k_final<<<M_BATCH, 96, 0, stream>>>(fsx, xs, pv, suma, esacc, out);
}